// GAT_1675037246077
// MI455X (gfx1250) — compile-verified
//
#include <hip/hip_runtime.h>
#include <math.h>

#define NNODES 100000
#define NEDGES 800000
#define ETOT   (NNODES + NEDGES)
#define SLOPE  0.2f

typedef __attribute__((ext_vector_type(2))) float v2f;
typedef __attribute__((ext_vector_type(8))) float v8f;

// ---------------- utility ----------------
__global__ void zero_u32(unsigned* __restrict__ p, int n) {
    int i = blockIdx.x * blockDim.x + threadIdx.x;
    if (i < n) p[i] = 0u;
}

// order-preserving float <-> uint encoding for atomicMax-based segment max
__device__ __forceinline__ unsigned encF(float f) {
    unsigned u = __float_as_uint(f);
    return (u & 0x80000000u) ? ~u : (u | 0x80000000u);
}
__device__ __forceinline__ float decF(unsigned e) {
    unsigned u = (e & 0x80000000u) ? (e & 0x7fffffffu) : ~e;
    return __uint_as_float(u);
}

// ---------------- fp32 WMMA GEMM: D[M,Nc] = A[M,K] @ B[K,Nc] (+ bias) ----------------
// one wave32 per 16x16 tile; M,Nc multiples of 16; K multiple of 4
__global__ void wmma_gemm_f32(const float* __restrict__ A, const float* __restrict__ B,
                              const float* __restrict__ bias, float* __restrict__ D,
                              int Nc, int K, int nTiles, int totalTiles) {
    int t = blockIdx.x * (blockDim.x >> 5) + (threadIdx.x >> 5);
    if (t >= totalTiles) return;                 // wave-uniform; EXEC all-1s inside
    const int mt = t / nTiles, nt = t % nTiles;
    const int row0 = mt << 4, col0 = nt << 4;
    const int lane = threadIdx.x & 31;
    const int half = lane >> 4, l16 = lane & 15;

    v8f acc = {};
    for (int k0 = 0; k0 < K; k0 += 4) {
        v2f a, b;
        // A 16x4 fragment: lanes 0-15 hold K = k0+0/1, lanes 16-31 hold K = k0+2/3
        const float* ap = A + (size_t)(row0 + l16) * K + (k0 + 2 * half);
        a.x = ap[0];
        a.y = ap[1];
        // B 4x16 fragment: VGPR0/1 hold rows k0+2*half, k0+2*half+1, column = l16
        const float* bp = B + (size_t)(k0 + 2 * half) * Nc + (col0 + l16);
        b.x = bp[0];
        b.y = bp[Nc];
        acc = __builtin_amdgcn_wmma_f32_16x16x4_f32(false, a, false, b,
                                                    (short)0, acc, false, false);
    }
    const float bv = bias ? bias[col0 + l16] : 0.0f;
#pragma unroll
    for (int v = 0; v < 8; ++v) {
        int row = row0 + v + 8 * half;           // C/D layout: VGPR v -> M = v (+8 for hi half)
        D[(size_t)row * Nc + (col0 + l16)] = acc[v] + bv;
    }
}

// ---------------- per-node attention logits ----------------
__global__ void alpha_kernel(const float* __restrict__ h, const float* __restrict__ a_s,
                             const float* __restrict__ a_d, float* __restrict__ als,
                             float* __restrict__ ald, int H, int C) {
    int n = blockIdx.x * blockDim.x + threadIdx.x;
    if (n >= NNODES) return;
    const float* hp = h + (size_t)n * H * C;
    for (int hh = 0; hh < H; ++hh) {
        float ss = 0.f, sd = 0.f;
        for (int c = 0; c < C; ++c) {
            float hv = hp[hh * C + c];
            ss += hv * a_s[hh * C + c];
            sd += hv * a_d[hh * C + c];
        }
        als[n * H + hh] = ss;
        ald[n * H + hh] = sd;
    }
}

// ---------------- edge pass 1: segment max of leaky_relu(e) over dst ----------------
__global__ void edge_max(const int* __restrict__ ei, const float* __restrict__ als,
                         const float* __restrict__ ald, unsigned* __restrict__ mEnc, int H) {
    int e = blockIdx.x * blockDim.x + threadIdx.x;
    if (e >= ETOT) return;
    int s = (e < NEDGES) ? ei[e] : (e - NEDGES);
    int d = (e < NEDGES) ? ei[NEDGES + e] : (e - NEDGES);
    for (int hh = 0; hh < H; ++hh) {
        float v = als[s * H + hh] + ald[d * H + hh];
        v = (v > 0.f) ? v : SLOPE * v;
        atomicMax(&mEnc[d * H + hh], encF(v));
    }
}

// ---------------- edge pass 2: p = exp(e - m[dst]); z[dst] += p ----------------
__global__ void edge_expsum(const int* __restrict__ ei, const float* __restrict__ als,
                            const float* __restrict__ ald, const unsigned* __restrict__ mEnc,
                            float* __restrict__ z, float* __restrict__ pBuf, int H) {
    int e = blockIdx.x * blockDim.x + threadIdx.x;
    if (e >= ETOT) return;
    int s = (e < NEDGES) ? ei[e] : (e - NEDGES);
    int d = (e < NEDGES) ? ei[NEDGES + e] : (e - NEDGES);
    for (int hh = 0; hh < H; ++hh) {
        float v = als[s * H + hh] + ald[d * H + hh];
        v = (v > 0.f) ? v : SLOPE * v;
        float p = expf(v - decF(mEnc[d * H + hh]));
        pBuf[(size_t)e * H + hh] = p;
        atomicAdd(&z[d * H + hh], p);
    }
}

// ---------------- edge pass 3: out[dst] += (p/z) * h[src]  (wave per edge) ----------------
__global__ void edge_aggregate(const int* __restrict__ ei, const float* __restrict__ hfeat,
                               const float* __restrict__ pBuf, const float* __restrict__ z,
                               float* __restrict__ out, int H, int C) {
    const int HC = H * C;
    int e = blockIdx.x * (blockDim.x >> 5) + (threadIdx.x >> 5);
    int lane = threadIdx.x & 31;
    if (e >= ETOT) return;
    int s = (e < NEDGES) ? ei[e] : (e - NEDGES);
    int d = (e < NEDGES) ? ei[NEDGES + e] : (e - NEDGES);
    const float* hs = hfeat + (size_t)s * HC;
    float* op = out + (size_t)d * HC;
    for (int c = lane; c < HC; c += 32) {
        int hh = c / C;
        float alpha = pBuf[(size_t)e * H + hh] / (z[d * H + hh] + 1e-16f);
        atomicAdd(&op[c], alpha * hs[c]);
    }
}

// ---------------- bias + SELU (in place) ----------------
__global__ void bias_selu(float* __restrict__ p, const float* __restrict__ bias,
                          int HC, int total) {
    int i = blockIdx.x * blockDim.x + threadIdx.x;
    if (i >= total) return;
    float v = p[i] + bias[i % HC];
    const float kScale = 1.0507009873554805f;
    const float kAlpha = 1.6732632423543772f;
    p[i] = (v > 0.f) ? kScale * v : kScale * kAlpha * (expf(v) - 1.f);
}

// ---------------- launch ----------------
extern "C" void kernel_launch(void* const* d_in, const int* in_sizes, int n_in,
                              void* d_out, int out_size, void* d_ws, size_t ws_size,
                              hipStream_t stream) {
    const float* x    = (const float*)d_in[0];
    const int*   ei   = (const int*)  d_in[1];
    const float* W1   = (const float*)d_in[2];
    const float* as1  = (const float*)d_in[3];
    const float* ad1  = (const float*)d_in[4];
    const float* b1   = (const float*)d_in[5];
    const float* W2   = (const float*)d_in[6];
    const float* as2  = (const float*)d_in[7];
    const float* ad2  = (const float*)d_in[8];
    const float* b2   = (const float*)d_in[9];
    const float* Wf   = (const float*)d_in[10];
    const float* bf   = (const float*)d_in[11];
    float* outp = (float*)d_out;

    // workspace layout (floats)
    float* ws = (float*)d_ws;
    size_t off = 0;
    float*    h1   = ws + off; off += (size_t)NNODES * 96;   // layer1 features; reused layer2
    float*    act1 = ws + off; off += (size_t)NNODES * 96;   // layer1 agg/SELU -> layer2 input
    float*    als  = ws + off; off += (size_t)NNODES * 3;
    float*    ald  = ws + off; off += (size_t)NNODES * 3;
    float*    zb   = ws + off; off += (size_t)NNODES * 3;
    unsigned* mEnc = (unsigned*)(ws + off); off += (size_t)NNODES * 3;
    float*    pBuf = ws + off; off += (size_t)ETOT * 3;
    float*    h2   = h1;                       // [N,32] layer2 features
    float*    agg2 = h1 + (size_t)NNODES * 32; // [N,32] layer2 aggregation

    const int TB = 256;
    const int edgeBlocks  = (ETOT + TB - 1) / TB;
    const int waveBlocks  = (ETOT + (TB / 32) - 1) / (TB / 32);
    const int nodeBlocks  = (NNODES + TB - 1) / TB;

    // ================= layer 1: GAT(16 -> 3 heads x 32) =================
    {
        int tiles = (NNODES / 16) * (96 / 16);
        wmma_gemm_f32<<<(tiles + 3) / 4, 128, 0, stream>>>(x, W1, nullptr, h1, 96, 16,
                                                           96 / 16, tiles);
        alpha_kernel<<<nodeBlocks, TB, 0, stream>>>(h1, as1, ad1, als, ald, 3, 32);
        zero_u32<<<(NNODES * 3 + TB - 1) / TB, TB, 0, stream>>>(mEnc, NNODES * 3);
        zero_u32<<<(NNODES * 3 + TB - 1) / TB, TB, 0, stream>>>((unsigned*)zb, NNODES * 3);
        zero_u32<<<(NNODES * 96 + TB - 1) / TB, TB, 0, stream>>>((unsigned*)act1, NNODES * 96);
        edge_max<<<edgeBlocks, TB, 0, stream>>>(ei, als, ald, mEnc, 3);
        edge_expsum<<<edgeBlocks, TB, 0, stream>>>(ei, als, ald, mEnc, zb, pBuf, 3);
        edge_aggregate<<<waveBlocks, TB, 0, stream>>>(ei, h1, pBuf, zb, act1, 3, 32);
        bias_selu<<<(NNODES * 96 + TB - 1) / TB, TB, 0, stream>>>(act1, b1, 96, NNODES * 96);
    }

    // ================= layer 2: GAT(96 -> 1 head x 32) =================
    {
        int tiles = (NNODES / 16) * (32 / 16);
        wmma_gemm_f32<<<(tiles + 3) / 4, 128, 0, stream>>>(act1, W2, nullptr, h2, 32, 96,
                                                           32 / 16, tiles);
        alpha_kernel<<<nodeBlocks, TB, 0, stream>>>(h2, as2, ad2, als, ald, 1, 32);
        zero_u32<<<(NNODES + TB - 1) / TB, TB, 0, stream>>>(mEnc, NNODES);
        zero_u32<<<(NNODES + TB - 1) / TB, TB, 0, stream>>>((unsigned*)zb, NNODES);
        zero_u32<<<(NNODES * 32 + TB - 1) / TB, TB, 0, stream>>>((unsigned*)agg2, NNODES * 32);
        edge_max<<<edgeBlocks, TB, 0, stream>>>(ei, als, ald, mEnc, 1);
        edge_expsum<<<edgeBlocks, TB, 0, stream>>>(ei, als, ald, mEnc, zb, pBuf, 1);
        edge_aggregate<<<waveBlocks, TB, 0, stream>>>(ei, h2, pBuf, zb, agg2, 1, 32);
        bias_selu<<<(NNODES * 32 + TB - 1) / TB, TB, 0, stream>>>(agg2, b2, 32, NNODES * 32);
    }

    // ================= final linear: [N,32] @ [32,16] + bf =================
    {
        int tiles = (NNODES / 16) * (16 / 16);
        wmma_gemm_f32<<<(tiles + 3) / 4, 128, 0, stream>>>(agg2, Wf, bf, outp, 16, 32,
                                                           16 / 16, tiles);
    }
}